// InfoNCELoss_10651518894748
// MI455X (gfx1250) — compile-verified
//
#include <hip/hip_runtime.h>
#include <hip/hip_bf16.h>

// ---------------------------------------------------------------------------
// InfoNCE loss, fused, MI455X (gfx1250, wave32).
//   pos: 2048x2048x128 GEMM of unit vectors -> exp -> row mean, fused via
//        V_WMMA_F32_16X16X4_F32. One wave per (16-row strip, 16-tile chunk),
//        A strip register-resident (64 VGPRs), 4 tiles in flight with
//        independent accumulator chains so B loads pipeline behind WMMA
//        issue; exp row-sums accumulate per lane; deterministic partial
//        stores (no atomics).
//   neg: 2048x1000 gathered length-128 dots, quad-per-negative, DS_SWIZZLE
//        butterfly reductions (1 DS op per step).
// ---------------------------------------------------------------------------

typedef __attribute__((ext_vector_type(2))) float v2f;
typedef __attribute__((ext_vector_type(8))) float v8f;

constexpr int N      = 2048;  // B*H*W = 2*32*32
constexpr int D      = 128;   // feature dim
constexpr int HW     = 1024;  // H*W
constexpr int KNEG   = 1000;  // negatives per pixel
constexpr int NTILES = N / 16;          // 128 tiles per dimension
constexpr int CHUNK  = 16;              // tn tiles per wave
constexpr int NCHUNK = NTILES / CHUNK;  // 8 column chunks
constexpr int MTILE  = 4;               // tiles in flight per wave

// DS_SWIZZLE group-of-32 xor-butterfly: offset = (xor_mask<<10) | and_mask(0x1f)
constexpr int SWZ_X1  = 0x041F;
constexpr int SWZ_X2  = 0x081F;
constexpr int SWZ_X4  = 0x101F;
constexpr int SWZ_X8  = 0x201F;
constexpr int SWZ_X16 = 0x401F;

template <int IMM>
__device__ __forceinline__ float swz_add(float v) {
  return v + __int_as_float(__builtin_amdgcn_ds_swizzle(__float_as_int(v), IMM));
}

// ---------------------------------------------------------------------------
// Kernel 1: transpose NCHW -> (N, D) row-major + L2 normalize both inputs.
// One wave per row; 4 elements/lane; single-DS-op butterfly reduction.
// ---------------------------------------------------------------------------
__global__ void nrm_kernel(const float* __restrict__ x1,
                           const float* __restrict__ x2,
                           float* __restrict__ x1n,
                           float* __restrict__ x2n) {
  const int lane = threadIdx.x & 31;
  const int wave = threadIdx.x >> 5;
  const int row  = blockIdx.x * 8 + wave;
  if (row >= N) return;

  const int  b    = row >> 10;
  const int  hw   = row & (HW - 1);
  const long base = (long)b * D * HW + hw;     // x[b, d, h, w] = base + d*HW

  float a0 = x1[base + (long)(lane      ) * HW];
  float a1 = x1[base + (long)(lane +  32) * HW];
  float a2 = x1[base + (long)(lane +  64) * HW];
  float a3 = x1[base + (long)(lane +  96) * HW];
  float s = a0 * a0 + a1 * a1 + a2 * a2 + a3 * a3;
  s = swz_add<SWZ_X1>(s); s = swz_add<SWZ_X2>(s); s = swz_add<SWZ_X4>(s);
  s = swz_add<SWZ_X8>(s); s = swz_add<SWZ_X16>(s);
  float r = 1.0f / fmaxf(sqrtf(s), 1e-12f);
  {
    float* o = x1n + row * D;
    o[lane] = a0 * r; o[lane + 32] = a1 * r;
    o[lane + 64] = a2 * r; o[lane + 96] = a3 * r;
  }

  float c0 = x2[base + (long)(lane      ) * HW];
  float c1 = x2[base + (long)(lane +  32) * HW];
  float c2 = x2[base + (long)(lane +  64) * HW];
  float c3 = x2[base + (long)(lane +  96) * HW];
  float t = c0 * c0 + c1 * c1 + c2 * c2 + c3 * c3;
  t = swz_add<SWZ_X1>(t); t = swz_add<SWZ_X2>(t); t = swz_add<SWZ_X4>(t);
  t = swz_add<SWZ_X8>(t); t = swz_add<SWZ_X16>(t);
  float q = 1.0f / fmaxf(sqrtf(t), 1e-12f);
  {
    float* o = x2n + row * D;
    o[lane] = c0 * q; o[lane + 32] = c1 * q;
    o[lane + 64] = c2 * q; o[lane + 96] = c3 * q;
  }
}

// ---------------------------------------------------------------------------
// Kernel 2: fused S = x1n @ x2n^T -> exp -> row partial sums.
// Wave = (tm strip, column chunk): 16 tiles of 16x16, K=128 via fp32 WMMAs,
// MTILE=4 tiles concurrently on independent accumulator chains.
//   A frag (16x4 f32): lane L holds A[L&15][2*(L>>4) + {0,1}]  (float2)
//   B frag (4x16 f32): symmetric layout of B^T = rows of x2n, same addressing.
//   C/D: elem e in lane L = S[16*tm + e + 8*(L>>4)][16*tn + (L&15)]
// ---------------------------------------------------------------------------
__global__ void pos_kernel(const float* __restrict__ x1n,
                           const float* __restrict__ x2n,
                           float* __restrict__ pos_part) {  // [NCHUNK][N]
  const int lane  = threadIdx.x & 31;
  const int wave  = threadIdx.x >> 5;
  const int wid   = blockIdx.x * 8 + wave;   // 0 .. 128*NCHUNK-1
  const int tm    = wid >> 3;                // 0..127
  const int chunk = wid & (NCHUNK - 1);      // 0..7
  const int rr    = lane & 15;
  const int hi    = lane >> 4;

  // Preload the full A strip: 32 K-steps of float2 (64 VGPRs), reused 16x.
  const float* ap = x1n + (tm * 16 + rr) * D + 2 * hi;
  v2f a[32];
#pragma unroll
  for (int i = 0; i < 32; ++i) a[i] = *(const v2f*)(ap + 4 * i);

  float esum[8] = {0.f, 0.f, 0.f, 0.f, 0.f, 0.f, 0.f, 0.f};

  for (int t = 0; t < CHUNK; t += MTILE) {
    // MTILE independent tiles -> MTILE independent WMMA dependency chains.
    const float* bp[MTILE];
#pragma unroll
    for (int j = 0; j < MTILE; ++j)
      bp[j] = x2n + ((chunk * CHUNK + t + j) * 16 + rr) * D + 2 * hi;

    v8f c[MTILE];
#pragma unroll
    for (int j = 0; j < MTILE; ++j)
      c[j] = (v8f){0.f, 0.f, 0.f, 0.f, 0.f, 0.f, 0.f, 0.f};

#pragma unroll
    for (int i = 0; i < 32; ++i) {
#pragma unroll
      for (int j = 0; j < MTILE; ++j) {
        v2f b = *(const v2f*)(bp[j] + 4 * i);
        // (neg_a, A, neg_b, B, c_mod, C, reuse_a, reuse_b)
        c[j] = __builtin_amdgcn_wmma_f32_16x16x4_f32(false, a[i], false, b,
                                                     (short)0, c[j],
                                                     false, false);
      }
    }

#pragma unroll
    for (int j = 0; j < MTILE; ++j)
#pragma unroll
      for (int e = 0; e < 8; ++e) esum[e] += __expf(c[j][e]);  // T == 1
  }

  // Sum over the 16 columns held by each 16-lane half; one store per row.
  float* out = pos_part + chunk * N + tm * 16 + 8 * hi;
#pragma unroll
  for (int e = 0; e < 8; ++e) {
    float v = esum[e];
    v = swz_add<SWZ_X1>(v); v = swz_add<SWZ_X2>(v);
    v = swz_add<SWZ_X4>(v); v = swz_add<SWZ_X8>(v);
    if (rr == 0) out[e] = v;
  }
}

// ---------------------------------------------------------------------------
// Kernel 3: neg[n] = mean_k exp( x1n[n] . x1n[idx[n,k]] ).  One wave / row.
// Quad of lanes per negative (8 negatives in flight); each lane dots a
// 32-element slice vs its preloaded own-row slice; 2-step intra-quad
// butterfly per k, 3-step cross-quad sum once at the end (disjoint k sets).
// ---------------------------------------------------------------------------
__global__ void neg_kernel(const float* __restrict__ x1n,
                           const int* __restrict__ idxs,
                           float* __restrict__ neg_mean) {
  const int lane  = threadIdx.x & 31;
  const int wave  = threadIdx.x >> 5;
  const int row   = blockIdx.x * 8 + wave;
  if (row >= N) return;
  const int qlane = lane & 3;   // slice within the 128-dim row
  const int quad  = lane >> 2;  // which negative of the group of 8

  float4 own[8];
#pragma unroll
  for (int i = 0; i < 8; ++i)
    own[i] = *(const float4*)(x1n + row * D + qlane * 32 + i * 4);

  const int* ip = idxs + row * KNEG;
  float acc = 0.0f;
  for (int k0 = 0; k0 < KNEG; k0 += 8) {
    __builtin_prefetch(ip + k0 + 64, 0, 1);     // global_prefetch_b8
    const int idx = ip[k0 + quad];
    const float* gp = x1n + idx * D + qlane * 32;
    float p = 0.0f;
#pragma unroll
    for (int i = 0; i < 8; ++i) {
      const float4 o = *(const float4*)(gp + i * 4);
      p += own[i].x * o.x + own[i].y * o.y + own[i].z * o.z + own[i].w * o.w;
    }
    p = swz_add<SWZ_X1>(p);     // intra-quad butterfly: all 4 lanes get dot
    p = swz_add<SWZ_X2>(p);
    acc += __expf(p);           // replicated across the 4 lanes of each quad
  }
  // xor 4/8/16 sums lanes with equal qlane -> disjoint k subsets: exact sum.
  acc = swz_add<SWZ_X4>(acc);
  acc = swz_add<SWZ_X8>(acc);
  acc = swz_add<SWZ_X16>(acc);
  if (lane == 0) neg_mean[row] = acc * (1.0f / KNEG);
}

// ---------------------------------------------------------------------------
// Kernel 4: loss = mean_n log((pos_mean + neg_mean) / pos_mean).
// Deterministic: sums the NCHUNK partials in fixed order.
// ---------------------------------------------------------------------------
__global__ void loss_kernel(const float* __restrict__ pos_part,
                            const float* __restrict__ neg_mean,
                            float* __restrict__ out) {
  __shared__ float red[256];
  float acc = 0.0f;
  for (int n = threadIdx.x; n < N; n += 256) {
    float ps = 0.0f;
#pragma unroll
    for (int ch = 0; ch < NCHUNK; ++ch) ps += pos_part[ch * N + n];
    const float p = ps * (1.0f / N);            // row sum -> row mean
    const float g = neg_mean[n];
    acc += logf((p + g) / p);                   // == -(log p - log(p+g))
  }
  red[threadIdx.x] = acc;
  __syncthreads();
  for (int s = 128; s > 0; s >>= 1) {
    if ((int)threadIdx.x < s) red[threadIdx.x] += red[threadIdx.x + s];
    __syncthreads();
  }
  if (threadIdx.x == 0) out[0] = red[0] * (1.0f / N);
}

// ---------------------------------------------------------------------------
extern "C" void kernel_launch(void* const* d_in, const int* in_sizes, int n_in,
                              void* d_out, int out_size, void* d_ws, size_t ws_size,
                              hipStream_t stream) {
  (void)in_sizes; (void)n_in; (void)out_size; (void)ws_size;

  const float* x1   = (const float*)d_in[0];
  const float* x2   = (const float*)d_in[1];
  const int*   idxs = (const int*)d_in[2];

  float* ws       = (float*)d_ws;
  float* x1n      = ws;                          // N*D floats
  float* x2n      = x1n + (size_t)N * D;         // N*D floats
  float* pos_part = x2n + (size_t)N * D;         // NCHUNK*N floats
  float* neg_mean = pos_part + (size_t)NCHUNK * N;  // N floats
  float* out      = (float*)d_out;

  nrm_kernel <<<N / 8, 256, 0, stream>>>(x1, x2, x1n, x2n);
  pos_kernel <<<(NTILES * NCHUNK) / 8, 256, 0, stream>>>(x1n, x2n, pos_part);
  neg_kernel <<<N / 8, 256, 0, stream>>>(x1n, idxs, neg_mean);
  loss_kernel<<<1, 256, 0, stream>>>(pos_part, neg_mean, out);
}